// Sn_coarsen_layer_6296422056037
// MI455X (gfx1250) — compile-verified
//
#include <hip/hip_runtime.h>
#include <hip/hip_bf16.h>

typedef __attribute__((ext_vector_type(2))) float v2f;
typedef __attribute__((ext_vector_type(8))) float v8f;

constexpr int B_    = 16;
constexpr int N_    = 65536;
constexpr int C_    = 256;
constexpr int DIN   = 16;
constexpr int DOUT  = 16;
constexpr int RATIO = N_ / C_;   // 256

// ---------------------------------------------------------------------------
// Pass 1: s[b,c,i] = sum over the 256 permuted rows of cluster c.
// One block per (b,c). Thread t: i = t&15, group g = t>>4 sums 16 rows.
// ---------------------------------------------------------------------------
__global__ __launch_bounds__(256)
void k_sum(const float* __restrict__ x, const int* __restrict__ pid,
           float* __restrict__ s) {
  __shared__ float red[16][17];
  int bc = blockIdx.x;
  int b = bc >> 8;
  int c = bc & 255;
  int t = threadIdx.x;
  int i = t & 15;
  int g = t >> 4;

  int jbase = c * RATIO + g * 16;
  float acc = 0.f;
#pragma unroll
  for (int k = 0; k < 16; ++k) {
    int n = pid[jbase + k];
    acc += x[((size_t)b * N_ + n) * DIN + i];
  }
  red[g][i] = acc;
  __syncthreads();
  if (t < 16) {
    float r = 0.f;
#pragma unroll
    for (int g2 = 0; g2 < 16; ++g2) r += red[g2][t];
    s[(b * C_ + c) * DIN + t] = r;
  }
}

// ---------------------------------------------------------------------------
// Pass 2: off[b,c,o] = b1[o] + (1/N) * sum_{i,k} w_off[o,i,c,k] * s[b,k,i]
// One block per c. Stage w_off slice and s slice in LDS (pitch 257 to avoid
// bank conflicts). Thread t -> (o = t>>4, b = t&15). Streams the 67MB w_off.
// ---------------------------------------------------------------------------
__global__ __launch_bounds__(256)
void k_off(const float* __restrict__ w_off, const float* __restrict__ s,
           const float* __restrict__ b1, float* __restrict__ off) {
  __shared__ float wsh[16][257];
  __shared__ float ssh[16][257];
  int c = blockIdx.x;
  int t = threadIdx.x;
  int o  = t >> 4;
  int bq = t & 15;

  float acc = 0.f;
  for (int i = 0; i < DIN; ++i) {
    __syncthreads();
#pragma unroll
    for (int kk = 0; kk < 16; ++kk) {
      int idx = t + 256 * kk;            // 0..4095
      int oo = idx >> 8;                 // 0..15
      int k  = idx & 255;
      wsh[oo][k] = w_off[(((size_t)oo * DIN + i) * C_ + c) * C_ + k]; // coalesced
      ssh[oo][k] = s[(oo * C_ + k) * DIN + i];                        // oo = b here
    }
    __syncthreads();
#pragma unroll 8
    for (int k = 0; k < C_; ++k)
      acc += wsh[o][k] * ssh[bq][k];
  }
  off[(bq * C_ + c) * DOUT + o] = b1[o] + acc * (1.0f / (float)N_);
}

// ---------------------------------------------------------------------------
// Pass 3: out[b, pid[j], :] = x[b, pid[j], :] @ w_diag[:,:,c]^T + off[b,c,:]
// One block per (b,c); 8 waves x 2 tiles of 16 rows. Each tile is a
// 16x16x16 GEMM done as 4 chained v_wmma_f32_16x16x4_f32.
//
// A layout (16x4 f32): lane L holds row M=L%16; VGPR p holds K = p + 2*(L/16)
// B layout (4x16 f32): lane L holds col N=L%16; VGPR p holds K = p + 2*(L/16)
// C/D layout (16x16 f32): lane L holds N=L%16, M = vgpr + 8*(L/16)
// ---------------------------------------------------------------------------
__global__ __launch_bounds__(256)
void k_out(const float* __restrict__ x, const float* __restrict__ w_diag,
           const float* __restrict__ off, const int* __restrict__ pid,
           float* __restrict__ out) {
  int bc = blockIdx.x;
  int b = bc >> 8;
  int c = bc & 255;
  int t = threadIdx.x;
  int wave   = t >> 5;
  int lane   = t & 31;
  int laneLo = lane & 15;
  int laneHi = lane >> 4;

  // B tile: w_diag[o = laneLo, i, c]; shared by all tiles of this block.
  v2f bt[4];
#pragma unroll
  for (int sl = 0; sl < 4; ++sl) {
    int i0 = 4 * sl + 2 * laneHi;
    bt[sl].x = w_diag[(laneLo * DIN + i0    ) * C_ + c];
    bt[sl].y = w_diag[(laneLo * DIN + i0 + 1) * C_ + c];
  }
  // accumulator seed: off[b,c,o] (bias + offset term), broadcast over M
  float offv = off[(b * C_ + c) * DOUT + laneLo];

#pragma unroll
  for (int tt = 0; tt < 2; ++tt) {
    int tile  = wave * 2 + tt;
    int jbase = c * RATIO + tile * 16;

    // A tile: row M = laneLo gathered through permute_ids
    int rowA = pid[jbase + laneLo];
    const float* xrow = x + ((size_t)b * N_ + rowA) * DIN + 2 * laneHi;
    v2f a[4];
#pragma unroll
    for (int sl = 0; sl < 4; ++sl)
      a[sl] = *(const v2f*)(xrow + 4 * sl);   // 8B-aligned (even float offset)

    v8f acc;
#pragma unroll
    for (int v = 0; v < 8; ++v) acc[v] = offv;

#pragma unroll
    for (int sl = 0; sl < 4; ++sl)
      acc = __builtin_amdgcn_wmma_f32_16x16x4_f32(
          /*neg_a=*/false, a[sl], /*neg_b=*/false, bt[sl],
          /*c_mod=*/(short)0, acc, /*reuse_a=*/false, /*reuse_b=*/false);

    // scatter D rows: lane writes element (M = v + 8*laneHi, N = laneLo)
#pragma unroll
    for (int v = 0; v < 8; ++v) {
      int m = v + 8 * laneHi;
      int n = pid[jbase + m];
      out[((size_t)b * N_ + n) * DOUT + laneLo] = acc[v];
    }
  }
}

extern "C" void kernel_launch(void* const* d_in, const int* in_sizes, int n_in,
                              void* d_out, int out_size, void* d_ws, size_t ws_size,
                              hipStream_t stream) {
  (void)in_sizes; (void)n_in; (void)out_size; (void)ws_size;
  const float* x      = (const float*)d_in[0];
  const float* w_diag = (const float*)d_in[1];
  const float* w_off  = (const float*)d_in[2];
  const float* b1     = (const float*)d_in[3];
  const int*   pid    = (const int*)d_in[4];
  float* out = (float*)d_out;

  float* s_ws   = (float*)d_ws;                 // B*C*DIN  = 65536 floats
  float* off_ws = s_ws + B_ * C_ * DIN;         // B*C*DOUT = 65536 floats

  k_sum<<<B_ * C_, 256, 0, stream>>>(x, pid, s_ws);
  k_off<<<C_, 256, 0, stream>>>(w_off, s_ws, b1, off_ws);
  k_out<<<B_ * C_, 256, 0, stream>>>(x, w_diag, off_ws, pid, out);
}